// CHIVE_29618094473726
// MI455X (gfx1250) — compile-verified
//
#include <hip/hip_runtime.h>
#include <math.h>

#define TT   262144
#define HH   32
#define NSYL 32768

typedef float v2f __attribute__((ext_vector_type(2)));
typedef float v8f __attribute__((ext_vector_type(8)));

#if __has_builtin(__builtin_amdgcn_tanhf)
#define TANHF(x) __builtin_amdgcn_tanhf(x)
#else
#define TANHF(x) tanhf(x)
#endif

static __device__ __forceinline__ v8f zero8() {
  v8f z;
#pragma unroll
  for (int i = 0; i < 8; ++i) z[i] = 0.f;
  return z;
}

// Full-precision CDNA5 WMMA: D(16x16 f32) = A(16x4 f32) * B(4x16 f32) + C
static __device__ __forceinline__ v8f wmma_k4(v2f a, v2f b, v8f c) {
  return __builtin_amdgcn_wmma_f32_16x16x4_f32(false, a, false, b, (short)0, c,
                                               false, false);
}

// Branch-free K-padding: clamp the index so the load is always in-bounds,
// then select the value (v_cndmask) -- avoids exec-mask branches + per-load
// waits that serialize the WMMA pipeline.
static __device__ __forceinline__ float wpad(const float* __restrict__ W,
                                             int k, int kmax, int col) {
  const int kk = (k < kmax) ? k : 0;
  const float v = W[(size_t)kk * HH + col];
  return (k < kmax) ? v : 0.f;
}
static __device__ __forceinline__ float apad(const float* __restrict__ A,
                                             int k, int kmax) {
  const int kk = (k < kmax) ? k : 0;
  const float v = A[kk];
  return (k < kmax) ? v : 0.f;
}

// ---------------------------------------------------------------------------
// Kernel 1: batched input projections with V_WMMA_F32_16X16X4_F32.
//   Xf[t][:] = frnn_seq[t] @ frnn_Wx + frnn_b          (K=13 padded to 16)
//   Xp[t][:] = phrnn_seq[t] @ phrnn_Wx + phrnn_b       (K=3  padded to 4)
//   Xs[k][:] = sylrnn_seq[syl_positions[k]] @ syl_Wx   (K=32)
// 256 threads = 8 waves; each wave owns one 16-row tile; B fragments are
// preloaded into registers once per wave and reused.
// A layout (16x4 f32): lane l -> M=l%16, K = 2*(l/16) + vgpr
// B layout (4x16 f32): lane l -> N=l%16, K = 2*(l/16) + vgpr
// C/D (16x16 f32):     lane l, vgpr v -> M = v + 8*(l/16), N = l%16
// ---------------------------------------------------------------------------
__global__ __launch_bounds__(256) void chive_proj(
    const float* __restrict__ frnn_seq, const float* __restrict__ phrnn_seq,
    const float* __restrict__ sylrnn_seq, const int* __restrict__ syl_pos,
    const float* __restrict__ frnn_Wx, const float* __restrict__ frnn_b,
    const float* __restrict__ phrnn_Wx, const float* __restrict__ phrnn_b,
    const float* __restrict__ syl_Wx,
    float* __restrict__ Xf, float* __restrict__ Xp, float* __restrict__ Xs) {
  const int l    = threadIdx.x & 31;
  const int wid  = threadIdx.x >> 5;
  const int m    = l & 15;
  const int hl   = l >> 4;
  const int n    = l & 15;
  const int tile = blockIdx.x * 8 + wid;
  const int r0   = tile * 16;

  // ---- preload all B (weight) fragments, branch-free padded
  v2f bf[2][4], bp[2], bs[2][8];
#pragma unroll
  for (int nt = 0; nt < 2; ++nt) {
    const int col = nt * 16 + n;
#pragma unroll
    for (int kc = 0; kc < 4; ++kc) {
      const int k0 = kc * 4 + 2 * hl;
      bf[nt][kc].x = wpad(frnn_Wx, k0, 13, col);
      bf[nt][kc].y = wpad(frnn_Wx, k0 + 1, 13, col);
    }
    bp[nt].x = wpad(phrnn_Wx, 2 * hl, 3, col);
    bp[nt].y = wpad(phrnn_Wx, 2 * hl + 1, 3, col);
#pragma unroll
    for (int kc = 0; kc < 8; ++kc) {
      const int k0 = kc * 4 + 2 * hl;
      bs[nt][kc].x = syl_Wx[(size_t)k0 * HH + col];
      bs[nt][kc].y = syl_Wx[(size_t)(k0 + 1) * HH + col];
    }
  }

  // ---- A fragments (hoisted so loads batch ahead of the WMMA chain)
  const float* Af = frnn_seq + (size_t)(r0 + m) * 13;
  v2f af[4];
#pragma unroll
  for (int kc = 0; kc < 4; ++kc) {
    const int k0 = kc * 4 + 2 * hl;
    af[kc].x = apad(Af, k0, 13);
    af[kc].y = apad(Af, k0 + 1, 13);
  }
  const float* Ap = phrnn_seq + (size_t)(r0 + m) * 3;
  v2f ap;
  ap.x = apad(Ap, 2 * hl, 3);
  ap.y = apad(Ap, 2 * hl + 1, 3);

  // ---- Xf / Xp tiles
#pragma unroll
  for (int nt = 0; nt < 2; ++nt) {
    v8f acc = zero8();
#pragma unroll
    for (int kc = 0; kc < 4; ++kc) acc = wmma_k4(af[kc], bf[nt][kc], acc);
    const v8f accp  = wmma_k4(ap, bp[nt], zero8());
    const float bfv = frnn_b[nt * 16 + n];
    const float bpv = phrnn_b[nt * 16 + n];
#pragma unroll
    for (int v = 0; v < 8; ++v) {
      const size_t row = (size_t)(r0 + v + 8 * hl);
      Xf[row * HH + nt * 16 + n] = acc[v] + bfv;
      Xp[row * HH + nt * 16 + n] = accp[v] + bpv;
    }
  }

  // ---- Xs: only the syllable-boundary rows (first 2048 tiles), K=32
  if (tile < (NSYL / 16)) {
    const int trow  = syl_pos[r0 + m];
    const float* As = sylrnn_seq + (size_t)trow * HH;
    v2f av[8];
#pragma unroll
    for (int kc = 0; kc < 8; ++kc) {
      const int k0 = kc * 4 + 2 * hl;
      av[kc].x = As[k0];
      av[kc].y = As[k0 + 1];
    }
#pragma unroll
    for (int nt = 0; nt < 2; ++nt) {
      v8f acc = zero8();
#pragma unroll
      for (int kc = 0; kc < 8; ++kc) acc = wmma_k4(av[kc], bs[nt][kc], acc);
#pragma unroll
      for (int v = 0; v < 8; ++v)
        Xs[(size_t)(r0 + v + 8 * hl) * HH + nt * 16 + n] = acc[v];
    }
  }
}

// ---------------------------------------------------------------------------
// Kernel 2: sequential encoder scan. Single wave32, lane j == hidden unit j.
// h vectors in LDS (broadcast reads), Wh columns in registers (4x32 VGPRs).
// Clock gates are wave-uniform; t % c (c in 1..8) via divisibility bitmask.
// Software-pipelined: next step's clocks/gates/Xf/Xp rows are fetched while
// the current step's (serial) tanh/matvec chain executes.
// ---------------------------------------------------------------------------
__global__ __launch_bounds__(32) void chive_enc_scan(
    const float* __restrict__ Xf, const float* __restrict__ Xp,
    const float* __restrict__ Xs, const int* __restrict__ fclk,
    const int* __restrict__ pclk, const int* __restrict__ sclk,
    const int* __restrict__ sfq, const float* __restrict__ Whf,
    const float* __restrict__ Whp, const float* __restrict__ Wxs,
    const float* __restrict__ Whs, const float* __restrict__ sylb,
    float* __restrict__ bott) {
  const int j = threadIdx.x;
  __shared__ float hf[HH], hp[HH], hs[HH];
  float whf[HH], whp[HH], wxs[HH], whs[HH];
#pragma unroll
  for (int k = 0; k < HH; ++k) {
    whf[k] = Whf[k * HH + j];
    whp[k] = Whp[k * HH + j];
    wxs[k] = Wxs[k * HH + j];
    whs[k] = Whs[k * HH + j];
  }
  hf[j] = 0.f; hp[j] = 0.f; hs[j] = 0.f;
  const float sb = sylb[j];
  __syncthreads();

  // pipeline prologue
  int cf = fclk[0], cp = pclk[0], cs = sclk[0], sf = sfq[0];
  float xf = Xf[j], xp = Xp[j];

  int fc = 0, m3 = 0, m5 = 0, m7 = 0;  // m3 = t%3, m5 = t%5, m7 = t%7
  for (int t = 0; t < TT; ++t) {
    // ---- prefetch step t+1 (clamped; overlaps the serial compute below)
    const int tn   = (t + 1 < TT) ? (t + 1) : t;
    const int cf_n = fclk[tn], cp_n = pclk[tn], cs_n = sclk[tn],
              sf_n = sfq[tn];
    const float xf_n = Xf[(size_t)tn * HH + j];
    const float xp_n = Xp[(size_t)tn * HH + j];

    // bit c-1 set iff t % c == 0, c in 1..8 (no integer division)
    unsigned mask = 1u;
    if (!(t & 1)) mask |= 0x02u;
    if (m3 == 0)  mask |= 0x04u;
    if (!(t & 3)) mask |= 0x08u;
    if (m5 == 0)  mask |= 0x10u;
    if (m3 == 0 && !(t & 1)) mask |= 0x20u;
    if (m7 == 0)  mask |= 0x40u;
    if (!(t & 7)) mask |= 0x80u;
    const bool upf = (mask >> (cf - 1)) & 1u;
    const bool upp = (mask >> (cp - 1)) & 1u;

    float nhf = 0.f, nhp = 0.f;
    if (upf) {
      float a = xf;
#pragma unroll
      for (int k = 0; k < HH; ++k) a += hf[k] * whf[k];
      nhf = TANHF(a);
    }
    if (upp) {
      float a = xp;
#pragma unroll
      for (int k = 0; k < HH; ++k) a += hp[k] * whp[k];
      nhp = TANHF(a);
    }
    if (upf || upp) {
      __syncthreads();
      if (upf) hf[j] = nhf;
      if (upp) hp[j] = nhp;
      __syncthreads();
    }

    if (sf) {
      if (fc % cs == 0) {
        float a = Xs[(size_t)fc * HH + j] + sb;
#pragma unroll
        for (int k = 0; k < HH; ++k)
          a += (hf[k] + hp[k]) * wxs[k] + hs[k] * whs[k];
        const float nhs = TANHF(a);
        __syncthreads();
        hs[j] = nhs;
        __syncthreads();
      }
      bott[(size_t)fc * HH + j] = hs[j];
      ++fc;
    }
    m3 = (m3 == 2) ? 0 : m3 + 1;
    m5 = (m5 == 4) ? 0 : m5 + 1;
    m7 = (m7 == 6) ? 0 : m7 + 1;
    cf = cf_n; cp = cp_n; cs = cs_n; sf = sf_n; xf = xf_n; xp = xp_n;
  }
}

// ---------------------------------------------------------------------------
// Kernel 3: decoder precompute. h_sd_n / h_pd_n have NO recurrence in the
// reference (hidden reset to zeros), so both 32x32 GEMMs run batched with
// WMMA, then the four downstream dot-products are precomputed per syllable.
// ---------------------------------------------------------------------------
__global__ __launch_bounds__(32) void chive_dec_pre(
    const float* __restrict__ bott, const int* __restrict__ dclk,
    const float* __restrict__ sylWx, const float* __restrict__ sylb,
    const float* __restrict__ phdWx, const float* __restrict__ phdb,
    const float* __restrict__ pdurWx, const float* __restrict__ pdurb,
    const float* __restrict__ sdurWx, const float* __restrict__ sdurb,
    const float* __restrict__ fcWx, const float* __restrict__ fcb,
    const float* __restrict__ ffWx, const float* __restrict__ ffb,
    float* __restrict__ a_pdur, float* __restrict__ a_sdur,
    float* __restrict__ a_ff, float* __restrict__ a_fc) {
  const int l  = threadIdx.x;
  const int m  = l & 15;
  const int hl = l >> 4;
  const int n  = l & 15;
  const int s0 = blockIdx.x * 16;
  __shared__ float sdn[16][HH + 1];
  __shared__ float pdn[16][HH + 1];

  // Hoist gate clocks for the 8 rows this lane-half covers.
  int cdv[8];
#pragma unroll
  for (int v = 0; v < 8; ++v) cdv[v] = dclk[s0 + v + 8 * hl];

  // GEMM1: sd_n = gate(i%cd==0) ? tanh(bott @ syl_Wx + syl_b) : 0
  const float* Ar = bott + (size_t)(s0 + m) * HH;
  v2f av[8];
#pragma unroll
  for (int kc = 0; kc < 8; ++kc) {
    const int k0 = kc * 4 + 2 * hl;
    av[kc].x = Ar[k0];
    av[kc].y = Ar[k0 + 1];
  }
#pragma unroll
  for (int nt = 0; nt < 2; ++nt) {
    v2f bv[8];
#pragma unroll
    for (int kc = 0; kc < 8; ++kc) {
      const int k0 = kc * 4 + 2 * hl;
      bv[kc].x = sylWx[(size_t)k0 * HH + nt * 16 + n];
      bv[kc].y = sylWx[(size_t)(k0 + 1) * HH + nt * 16 + n];
    }
    v8f acc = zero8();
#pragma unroll
    for (int kc = 0; kc < 8; ++kc) acc = wmma_k4(av[kc], bv[kc], acc);
    const float bias = sylb[nt * 16 + n];
#pragma unroll
    for (int v = 0; v < 8; ++v) {
      const int row  = v + 8 * hl;
      const bool gat = ((s0 + row) % cdv[v]) == 0;
      sdn[row][nt * 16 + n] = gat ? TANHF(acc[v] + bias) : 0.f;
    }
  }
  __syncthreads();

  // GEMM2: pd_n = gate ? tanh(sd_n @ phdecd_Wx + phdecd_b) : 0
#pragma unroll
  for (int kc = 0; kc < 8; ++kc) {
    const int k0 = kc * 4 + 2 * hl;
    av[kc].x = sdn[m][k0];
    av[kc].y = sdn[m][k0 + 1];
  }
#pragma unroll
  for (int nt = 0; nt < 2; ++nt) {
    v2f bv[8];
#pragma unroll
    for (int kc = 0; kc < 8; ++kc) {
      const int k0 = kc * 4 + 2 * hl;
      bv[kc].x = phdWx[(size_t)k0 * HH + nt * 16 + n];
      bv[kc].y = phdWx[(size_t)(k0 + 1) * HH + nt * 16 + n];
    }
    v8f acc = zero8();
#pragma unroll
    for (int kc = 0; kc < 8; ++kc) acc = wmma_k4(av[kc], bv[kc], acc);
    const float bias = phdb[nt * 16 + n];
#pragma unroll
    for (int v = 0; v < 8; ++v) {
      const int row  = v + 8 * hl;
      const bool gat = ((s0 + row) % cdv[v]) == 0;
      pdn[row][nt * 16 + n] = gat ? TANHF(acc[v] + bias) : 0.f;
    }
  }
  __syncthreads();

  // Per-syllable dot products feeding the scalar decoder recurrences.
  const int s = s0 + m;
  if (hl == 0) {
    float dp = pdurb[0], dff = ffb[0];
#pragma unroll
    for (int k = 0; k < HH; ++k) {
      dp  += pdn[m][k] * pdurWx[k];
      dff += pdn[m][k] * ffWx[k];
    }
    a_pdur[s] = dp;
    a_ff[s]   = dff;
    for (int jj = 0; jj < 6; ++jj) {
      float d = fcb[jj];
#pragma unroll
      for (int k = 0; k < HH; ++k) d += pdn[m][k] * fcWx[k * 12 + jj];
      a_fc[(size_t)s * 12 + jj] = d;
    }
  } else {
    float dsv = sdurb[0];
#pragma unroll
    for (int k = 0; k < HH; ++k) dsv += sdn[m][k] * sdurWx[k];
    a_sdur[s] = dsv;
    for (int jj = 6; jj < 12; ++jj) {
      float d = fcb[jj];
#pragma unroll
      for (int k = 0; k < HH; ++k) d += pdn[m][k] * fcWx[k * 12 + jj];
      a_fc[(size_t)s * 12 + jj] = d;
    }
  }
}

// ---------------------------------------------------------------------------
// Kernel 4: sequential decoder scan over precomputed activations.
// Only scalar recurrences (h_pdur, h_sdur, h_ff) plus a 12-wide h_fc.
// ---------------------------------------------------------------------------
__global__ __launch_bounds__(32) void chive_dec_scan(
    const float* __restrict__ a_pdur, const float* __restrict__ a_sdur,
    const float* __restrict__ a_ff, const float* __restrict__ a_fc,
    const int* __restrict__ sfq, const int* __restrict__ dcc,
    const float* __restrict__ fcWh, const float* __restrict__ fcb,
    const float* __restrict__ ffWh, const float* __restrict__ ffb,
    const float* __restrict__ pdurWh, const float* __restrict__ sdurWh,
    float* __restrict__ out) {
  const int j = threadIdx.x;
  __shared__ float hfc[12];
  float wfc[12];
#pragma unroll
  for (int k = 0; k < 12; ++k) wfc[k] = (j < 12) ? fcWh[k * 12 + j] : 0.f;
  if (j < 12) hfc[j] = 0.f;
  float h_ff = 0.f, h_pdur = 0.f, h_sdur = 0.f;
  float held_ff = ffb[0];                     // h_pd starts at zeros
  float held_fc = (j < 12) ? fcb[j] : 0.f;
  const float wph = pdurWh[0], wsh = sdurWh[0], wffh = ffWh[0];
  __syncthreads();

  // pipeline prologue
  int sfp = 0, cc = dcc[0];
  for (int i = 0; i < NSYL; ++i) {
    const int in    = (i + 1 < NSYL) ? (i + 1) : i;
    const int sfp_n = sfq[i];        // sample_freq[(i+1)-1]
    const int cc_n  = dcc[in];

    const bool cond = (i > 0) && ((sfp == 1) || (h_sdur == 1.0f));
    if (cond) {
      h_pdur  = TANHF(a_pdur[i] + h_pdur * wph);
      h_sdur  = TANHF(a_sdur[i] + h_sdur * wsh);
      held_ff = a_ff[i];
      if (j < 12) held_fc = a_fc[(size_t)i * 12 + j];
    }
    h_ff = TANHF(held_ff + h_ff * wffh);
    if (i % cc == 0) {
      float a = held_fc;
#pragma unroll
      for (int k = 0; k < 12; ++k) a += hfc[k] * wfc[k];
      const float nv = TANHF(a);
      __syncthreads();
      if (j < 12) hfc[j] = nv;
      __syncthreads();
    }
    sfp = sfp_n; cc = cc_n;
  }
  // Outputs flat in return order: h_ff[1], h_fc[12], h_pdur[1], h_sdur[1]
  if (j == 0) { out[0] = h_ff; out[13] = h_pdur; out[14] = h_sdur; }
  if (j < 12) out[1 + j] = hfc[j];
}

extern "C" void kernel_launch(void* const* d_in, const int* in_sizes, int n_in,
                              void* d_out, int out_size, void* d_ws,
                              size_t ws_size, hipStream_t stream) {
  (void)in_sizes; (void)n_in; (void)out_size; (void)ws_size;
  const float* frnn_seq   = (const float*)d_in[0];
  const float* phrnn_seq  = (const float*)d_in[1];
  const float* sylrnn_seq = (const float*)d_in[2];
  const int* frnn_clock   = (const int*)d_in[3];
  const int* phrnn_clock  = (const int*)d_in[4];
  const int* syl_clock    = (const int*)d_in[5];
  const int* sample_freq  = (const int*)d_in[6];
  const int* syl_positions = (const int*)d_in[7];
  const int* decd_clock   = (const int*)d_in[8];
  const int* decd_clock_c = (const int*)d_in[9];
  const float* frnn_Wx = (const float*)d_in[10];
  const float* frnn_Wh = (const float*)d_in[11];
  const float* frnn_b  = (const float*)d_in[12];
  const float* phrnn_Wx = (const float*)d_in[13];
  const float* phrnn_Wh = (const float*)d_in[14];
  const float* phrnn_b  = (const float*)d_in[15];
  const float* syl_Wx = (const float*)d_in[16];
  const float* syl_Wh = (const float*)d_in[17];
  const float* syl_b  = (const float*)d_in[18];
  const float* phdecd_Wx = (const float*)d_in[19];
  const float* phdecd_b  = (const float*)d_in[21];
  const float* phdur_Wx = (const float*)d_in[22];
  const float* phdur_Wh = (const float*)d_in[23];
  const float* phdur_b  = (const float*)d_in[24];
  const float* sdur_Wx = (const float*)d_in[25];
  const float* sdur_Wh = (const float*)d_in[26];
  const float* sdur_b  = (const float*)d_in[27];
  const float* fc_Wx = (const float*)d_in[28];
  const float* fc_Wh = (const float*)d_in[29];
  const float* fc_b  = (const float*)d_in[30];
  const float* ff_Wx = (const float*)d_in[31];
  const float* ff_Wh = (const float*)d_in[32];
  const float* ff_b  = (const float*)d_in[33];

  // Workspace carve-up (floats): Xf[T*32] Xp[T*32] Xs[NSYL*32]
  // bott[NSYL*32] a_pdur[NSYL] a_sdur[NSYL] a_ff[NSYL] a_fc[NSYL*12]
  float* Xf     = (float*)d_ws;
  float* Xp     = Xf + (size_t)TT * HH;
  float* Xs     = Xp + (size_t)TT * HH;
  float* bott   = Xs + (size_t)NSYL * HH;
  float* a_pdur = bott + (size_t)NSYL * HH;
  float* a_sdur = a_pdur + NSYL;
  float* a_ff   = a_sdur + NSYL;
  float* a_fc   = a_ff + NSYL;

  chive_proj<<<TT / 16 / 8, 256, 0, stream>>>(
      frnn_seq, phrnn_seq, sylrnn_seq, syl_positions, frnn_Wx, frnn_b,
      phrnn_Wx, phrnn_b, syl_Wx, Xf, Xp, Xs);
  chive_enc_scan<<<1, 32, 0, stream>>>(Xf, Xp, Xs, frnn_clock, phrnn_clock,
                                       syl_clock, sample_freq, frnn_Wh,
                                       phrnn_Wh, syl_Wx, syl_Wh, syl_b, bott);
  chive_dec_pre<<<NSYL / 16, 32, 0, stream>>>(
      bott, decd_clock, syl_Wx, syl_b, phdecd_Wx, phdecd_b, phdur_Wx, phdur_b,
      sdur_Wx, sdur_b, fc_Wx, fc_b, ff_Wx, ff_b, a_pdur, a_sdur, a_ff, a_fc);
  chive_dec_scan<<<1, 32, 0, stream>>>(a_pdur, a_sdur, a_ff, a_fc, sample_freq,
                                       decd_clock_c, fc_Wh, fc_b, ff_Wh, ff_b,
                                       phdur_Wh, sdur_Wh, (float*)d_out);
}